// MultiChannelSpikingAttention_54485955117492
// MI455X (gfx1250) — compile-verified
//
#include <hip/hip_runtime.h>
#include <hip/hip_bf16.h>
#include <stdint.h>

#define B_ROWS 512
#define T_LEN  8192
#define K_WIN  5
#define ROWS_PER_BLK 32
#define CHUNK 256          // t-span of stored (non-warmup) work per block
#define WARM  64           // warm-up steps (0.7^64 ~ 1e-10, < fp32 eps)
#define TILE  64           // t-span per TDM tile
#define NTILES 5           // 1 warmup tile + 4 main tiles
#define LROW  65           // padded LDS row stride in dwords (64 data + 1 TDM pad)

typedef unsigned int       u32;
typedef unsigned long long u64;
typedef u32 v4u_t __attribute__((ext_vector_type(4)));
typedef int v4i_t __attribute__((ext_vector_type(4)));
typedef int v8i_t __attribute__((ext_vector_type(8)));

#if defined(__has_builtin)
#if __has_builtin(__builtin_amdgcn_tensor_load_to_lds)
#define HAVE_TDM 1
#endif
#endif
#ifndef HAVE_TDM
#define HAVE_TDM 0
#endif

// ---------------------------------------------------------------------------
// TDM issue: 2D tile (TILE x ROWS_PER_BLK fp32) global -> LDS, with LDS
// padding of 1 dword per 64 dwords so each 64-elem row lands at stride 65
// (bank-conflict-free for lane-per-row readers).
// ---------------------------------------------------------------------------
__device__ __forceinline__ void tdm_issue_tile(const float* gsrc, u32 lds_off_bytes)
{
#if HAVE_TDM
    u64 ga = (u64)(uintptr_t)gsrc;
    v4u_t g0;
    g0[0] = 1u;                                   // count=1 valid descriptor
    g0[1] = lds_off_bytes;                        // lds_addr
    g0[2] = (u32)ga;                              // global_addr[31:0]
    g0[3] = (u32)((ga >> 32) & 0x01ffffffu)       // global_addr[56:32]
          | (2u << 30);                           // type=2 ("image")
    v8i_t g1;
    // data_size=2 (4B), pad_enable=1, pad_interval=5 (64 dwords), pad_amount=0 (1 dword)
    g1[0] = (int)((2u << 16) | (1u << 20) | (5u << 22));
    const u32 td0 = (u32)T_LEN;                   // tensor_dim0
    const u32 td1 = (u32)B_ROWS;                  // tensor_dim1
    g1[1] = (int)((td0 & 0xffffu) << 16);         // [63:48] = tensor_dim0 lo16
    g1[2] = (int)((td0 >> 16) | ((td1 & 0xffffu) << 16));
    g1[3] = (int)((td1 >> 16) | ((u32)TILE << 16));          // tile_dim0 = 64
    g1[4] = (int)((u32)ROWS_PER_BLK);                         // tile_dim1 = 32
    g1[5] = (int)((u32)T_LEN);                   // tensor_dim0_stride = 8192
    g1[6] = 0;
    g1[7] = 0;
    v4i_t gz = {0, 0, 0, 0};
#if __clang_major__ >= 23
    v8i_t gz8 = {0, 0, 0, 0, 0, 0, 0, 0};
    __builtin_amdgcn_tensor_load_to_lds(g0, g1, gz, gz, gz8, 0);
#else
    __builtin_amdgcn_tensor_load_to_lds(g0, g1, gz, gz, 0);
#endif
#else
    (void)gsrc; (void)lds_off_bytes;
#endif
}

// ---------------------------------------------------------------------------
// Kernel A: LIF spike scan + weighted channel sum -> raw sal.
// Grid (16, 32): x = row group of 32, y = 256-wide t chunk. Block = 1 wave32.
// Each lane scans one row serially over [h*256-64, h*256+256), double-
// buffered TDM tiles feeding bank-conflict-free LDS reads.
// ---------------------------------------------------------------------------
__global__ __launch_bounds__(32)
void spike_scan_kernel(const float* __restrict__ amp,
                       const float* __restrict__ pitch,
                       const float* __restrict__ boundary,
                       const float* __restrict__ decay,
                       const float* __restrict__ weights,
                       float* __restrict__ sal_out)
{
    __shared__ float ldsIn[2][3][ROWS_PER_BLK * LROW];  // 48.8 KB
    __shared__ float ldsSal[ROWS_PER_BLK * LROW];       //  8.1 KB

    const int lane = threadIdx.x;
    const int g    = blockIdx.x;            // row group
    const int h    = blockIdx.y;            // t chunk
    const int row0 = g * ROWS_PER_BLK;

    const float d0 = decay[0], d1 = decay[1], d2 = decay[2];
    const float w0 = weights[0], w1 = weights[1], w2 = weights[2];
    const float* in[3] = { amp, pitch, boundary };

    auto stage = [&](int tile, int buf) {
        const int t0 = h * CHUNK + (tile - 1) * TILE;   // tile 0 => warmup start
        for (int c = 0; c < 3; ++c) {
            const float* src = in[c] + (size_t)row0 * T_LEN + t0;
#if HAVE_TDM
            tdm_issue_tile(src, (u32)(uintptr_t)&ldsIn[buf][c][0]);
#else
            for (int j = lane; j < ROWS_PER_BLK * TILE; j += 32) {
                const int r = j >> 6, x = j & 63;
                ldsIn[buf][c][r * LROW + x] = src[(size_t)r * T_LEN + x];
            }
#endif
        }
    };

    const int firstTile = (h == 0) ? 1 : 0;             // row start: v=0 is exact
    stage(firstTile, firstTile & 1);

    float v0 = 0.0f, v1 = 0.0f, v2 = 0.0f;

    for (int tile = firstTile; tile < NTILES; ++tile) {
        const int buf = tile & 1;
        if (tile + 1 < NTILES) stage(tile + 1, (tile + 1) & 1);
#if HAVE_TDM
        if (tile + 1 < NTILES) __builtin_amdgcn_s_wait_tensorcnt(3);
        else                   __builtin_amdgcn_s_wait_tensorcnt(0);
        asm volatile("" ::: "memory");
#else
        __syncthreads();
#endif
        const float* A  = &ldsIn[buf][0][lane * LROW];
        const float* P  = &ldsIn[buf][1][lane * LROW];
        const float* Bd = &ldsIn[buf][2][lane * LROW];
        const bool main_tile = (tile >= 1);

        #pragma unroll 8
        for (int i = 0; i < TILE; ++i) {
            v0 = fmaf(d0, v0, A[i]);
            v1 = fmaf(d1, v1, P[i]);
            v2 = fmaf(d2, v2, Bd[i]);
            float s0 = (v0 >= 1.0f) ? 1.0f : 0.0f; v0 -= s0;
            float s1 = (v1 >= 1.0f) ? 1.0f : 0.0f; v1 -= s1;
            float s2 = (v2 >= 1.0f) ? 1.0f : 0.0f; v2 -= s2;
            if (main_tile) ldsSal[lane * LROW + i] = w0 * s0 + w1 * s1 + w2 * s2;
        }

        if (main_tile) {
            __syncthreads();   // single wave: lowers to dscnt wait / nop
            const int tbase = h * CHUNK + (tile - 1) * TILE;
            for (int rr = 0; rr < ROWS_PER_BLK; ++rr) {   // coalesced writeout
                const size_t base = (size_t)(row0 + rr) * T_LEN + tbase;
                sal_out[base + lane]      = ldsSal[rr * LROW + lane];
                sal_out[base + 32 + lane] = ldsSal[rr * LROW + 32 + lane];
            }
            __syncthreads();
        }
    }
}

// ---------------------------------------------------------------------------
// Kernel B: per-row stable top-5 (tie -> lowest index, matching lax.top_k),
// gain mu = 0.5 + 2*tanh(1.8 * mean(top5/(max+1e-6))), and fused in-place
// normalization of sal (monotone scaling => indices identical pre/post norm).
// ---------------------------------------------------------------------------
__global__ __launch_bounds__(256)
void topk_norm_kernel(float* __restrict__ sal,
                      float* __restrict__ mu_out,
                      float* __restrict__ idx_out)
{
    __shared__ float hv[256]; __shared__ int hi[256];
    __shared__ float rv[256]; __shared__ int ri[256]; __shared__ int rt[256];

    const int tid = threadIdx.x;
    const int row = blockIdx.x;
    float* srow = sal + (size_t)row * T_LEN;
    float4* srow4 = reinterpret_cast<float4*>(srow);

    float bv[K_WIN]; int bi[K_WIN];
    #pragma unroll
    for (int k = 0; k < K_WIN; ++k) { bv[k] = -1.0f; bi[k] = 0x7fffffff; }

    auto insert = [&](float v, int t) {          // strict > keeps stability
        if (v > bv[K_WIN - 1]) {
            int k = K_WIN - 1;
            while (k > 0 && v > bv[k - 1]) { bv[k] = bv[k - 1]; bi[k] = bi[k - 1]; --k; }
            bv[k] = v; bi[k] = t;
        }
    };

    for (int q = tid; q < T_LEN / 4; q += 256) {  // per-thread t ascending
        const float4 v = srow4[q];
        const int t = q * 4;
        insert(v.x, t); insert(v.y, t + 1); insert(v.z, t + 2); insert(v.w, t + 3);
    }

    hv[tid] = bv[0]; hi[tid] = bi[0];
    int pos = 0;
    float topv[K_WIN]; int topi[K_WIN];
    __syncthreads();

    for (int k = 0; k < K_WIN; ++k) {
        rv[tid] = hv[tid]; ri[tid] = hi[tid]; rt[tid] = tid;
        __syncthreads();
        for (int off = 128; off > 0; off >>= 1) {
            if (tid < off) {
                const float ov = rv[tid + off]; const int oi = ri[tid + off];
                const int   ot = rt[tid + off];
                if (ov > rv[tid] || (ov == rv[tid] && oi < ri[tid])) {
                    rv[tid] = ov; ri[tid] = oi; rt[tid] = ot;
                }
            }
            __syncthreads();
        }
        const float wv = rv[0]; const int wi = ri[0]; const int wt = rt[0];
        __syncthreads();
        topv[k] = wv; topi[k] = wi;
        if (tid == wt) {                       // winner advances its sorted head
            ++pos;
            hv[tid] = (pos < K_WIN) ? bv[pos] : -2.0f;
            hi[tid] = (pos < K_WIN) ? bi[pos] : 0x7fffffff;
        }
        __syncthreads();
    }

    // All threads hold the same topv/topi -> broadcast-free epilogue.
    const float inv = 1.0f / (topv[0] + 1e-6f);

    if (tid == 0) {
        const float avg = (topv[0] + topv[1] + topv[2] + topv[3] + topv[4]) * inv * 0.2f;
        mu_out[row] = 0.5f + 2.0f * tanhf(1.8f * avg);
        #pragma unroll
        for (int k = 0; k < K_WIN; ++k) idx_out[row * K_WIN + k] = (float)topi[k];
    }

    // Fused in-place normalization (raw sal fully read above; barriers passed).
    for (int q = tid; q < T_LEN / 4; q += 256) {
        float4 v = srow4[q];
        v.x *= inv; v.y *= inv; v.z *= inv; v.w *= inv;
        srow4[q] = v;
    }
}

// ---------------------------------------------------------------------------
extern "C" void kernel_launch(void* const* d_in, const int* in_sizes, int n_in,
                              void* d_out, int out_size, void* d_ws, size_t ws_size,
                              hipStream_t stream)
{
    (void)in_sizes; (void)n_in; (void)out_size; (void)d_ws; (void)ws_size;
    const float* amp      = (const float*)d_in[0];
    const float* pitch    = (const float*)d_in[1];
    const float* boundary = (const float*)d_in[2];
    const float* decay    = (const float*)d_in[3];
    const float* weights  = (const float*)d_in[4];

    float* out = (float*)d_out;
    float* mu  = out;                                     // [512]
    float* sal = out + B_ROWS;                            // [512*8192]
    float* idx = out + B_ROWS + (size_t)B_ROWS * T_LEN;   // [512*5]

    dim3 gridA(B_ROWS / ROWS_PER_BLK, T_LEN / CHUNK);     // 16 x 32 blocks
    spike_scan_kernel<<<gridA, 32, 0, stream>>>(amp, pitch, boundary, decay,
                                                weights, sal);
    topk_norm_kernel<<<B_ROWS, 256, 0, stream>>>(sal, mu, idx);
}